// CATCH_86294482911873
// MI455X (gfx1250) — compile-verified
//
#include <hip/hip_runtime.h>
#include <hip/hip_bf16.h>

typedef __attribute__((ext_vector_type(2))) float v2f;
typedef __attribute__((ext_vector_type(8))) float v8f;
typedef int v4i_ __attribute__((vector_size(16)));

#define B_  16
#define L_  1024
#define C_  32
#define P_  16
#define S_  8
#define PN_ 127
#define H_  256
#define NH_ 8
#define HD_ 32
#define NL_ 2
#define FH_ 1024   // FR*H

// ---------------- async global->LDS staging (CDNA5), guarded ----------------
#if defined(__has_builtin)
#if __has_builtin(__builtin_amdgcn_global_load_async_to_lds_b128)
#define HAVE_ASYNC_LDS 1
#endif
#endif

#ifdef HAVE_ASYNC_LDS
__device__ __forceinline__ void async_cp16(const float* g, float* l) {
  __builtin_amdgcn_global_load_async_to_lds_b128(
      (v4i_*)(void*)const_cast<float*>(g), (v4i_*)(void*)l, 0, 0);
}
__device__ __forceinline__ void async_wait0() {
#if __has_builtin(__builtin_amdgcn_s_wait_asynccnt)
  __builtin_amdgcn_s_wait_asynccnt(0);
#else
  asm volatile("s_wait_asynccnt 0x0" ::: "memory");
#endif
}
#endif

// Stage C_*H_ floats (32KB tile) from global to LDS (all 256 threads).
__device__ __forceinline__ void stage_tile_8k(const float* __restrict__ gsrc,
                                              float* __restrict__ ldst, int tid) {
#ifdef HAVE_ASYNC_LDS
  for (int e = tid * 4; e < C_ * H_; e += 1024) async_cp16(gsrc + e, ldst + e);
  async_wait0();
#else
  for (int e = tid; e < C_ * H_; e += 256) ldst[e] = gsrc[e];
#endif
}

// ---------------- WMMA helpers (fp32 16x16x4) ----------------
__device__ __forceinline__ v8f wmma4(v2f a, v2f b, v8f c) {
  return __builtin_amdgcn_wmma_f32_16x16x4_f32(false, a, false, b, (short)0, c, false, false);
}

// D = A(16xK) * B(Kx16) + acc.  A row-major lda, B row-major ldb (B[k*ldb+n]).
__device__ __forceinline__ v8f tile_gemm(const float* __restrict__ A, int lda,
                                         const float* __restrict__ Bp, int ldb,
                                         int K, v8f acc) {
  const int lane = threadIdx.x & 31;
  const float* a  = A  + (lane & 15) * lda + ((lane >> 4) << 1);
  const float* bq = Bp + (lane & 15) + (long)((lane >> 4) << 1) * ldb;
  for (int k = 0; k < K; k += 4) {
    v2f av = { a[k], a[k + 1] };
    v2f bv = { bq[(long)k * ldb], bq[(long)(k + 1) * ldb] };
    acc = wmma4(av, bv, acc);
  }
  return acc;
}

// Same but B supplied transposed: element (k,n) = Bt[n*ldbt + k]
__device__ __forceinline__ v8f tile_gemm_bt(const float* __restrict__ A, int lda,
                                            const float* __restrict__ Bt, int ldbt,
                                            int K, v8f acc) {
  const int lane = threadIdx.x & 31;
  const float* a  = A  + (lane & 15) * lda + ((lane >> 4) << 1);
  const float* bq = Bt + (long)(lane & 15) * ldbt + ((lane >> 4) << 1);
  for (int k = 0; k < K; k += 4) {
    v2f av = { a[k], a[k + 1] };
    v2f bv = { bq[k], bq[k + 1] };
    acc = wmma4(av, bv, acc);
  }
  return acc;
}

__device__ __forceinline__ float hash_u(unsigned a, unsigned b) {
  unsigned x = a * 0x9E3779B9u + b * 0x85EBCA6Bu + 0x27220A95u;
  x ^= x >> 16; x *= 0x7FEB352Du; x ^= x >> 15; x *= 0x846CA68Bu; x ^= x >> 16;
  return ((x >> 8) + 0.5f) * (1.0f / 16777216.0f);
}

__device__ __forceinline__ float geluf(float x) {
  return 0.5f * x * (1.0f + erff(x * 0.70710678118f));
}

// ---------------- 1) LayerNorm over time per (b,c) ----------------
__global__ void k_ln_time(const float* __restrict__ x, const float* __restrict__ w,
                          const float* __restrict__ bb, float* __restrict__ xn) {
  int bc = blockIdx.x; int b = bc >> 5; int c = bc & 31;
  const float* xp = x + (long)b * L_ * C_ + c;
  __shared__ float rs[256], rq[256];
  float s = 0.f, q = 0.f;
  for (int l = threadIdx.x; l < L_; l += 256) { float v = xp[(long)l * C_]; s += v; q += v * v; }
  rs[threadIdx.x] = s; rq[threadIdx.x] = q; __syncthreads();
  for (int off = 128; off > 0; off >>= 1) {
    if (threadIdx.x < off) { rs[threadIdx.x] += rs[threadIdx.x + off]; rq[threadIdx.x] += rq[threadIdx.x + off]; }
    __syncthreads();
  }
  float mean = rs[0] * (1.0f / L_);
  float var  = rq[0] * (1.0f / L_) - mean * mean;
  float rstd = rsqrtf(var + 1e-5f);
  float ww = w[c], vb = bb[c];
  float* o = xn + (long)bc * L_;
  for (int l = threadIdx.x; l < L_; l += 256)
    o[l] = (xp[(long)l * C_] - mean) * rstd * ww + vb;
}

// ---------------- 2) Forward FFT (radix-2 DIT, 1024-pt, per (b,c)) ----------------
__global__ void k_fft_fwd(const float* __restrict__ xn, float* __restrict__ xr,
                          float* __restrict__ xi) {
  __shared__ float re[1024], im[1024];
  long base = (long)blockIdx.x * L_;
  for (int i = threadIdx.x; i < 1024; i += 256) {
    int j = __brev((unsigned)i) >> 22;
    re[j] = xn[base + i]; im[j] = 0.f;
  }
  __syncthreads();
  for (int len = 2; len <= 1024; len <<= 1) {
    int half = len >> 1;
    float ang0 = -6.283185307179586f / (float)len;
    for (int t = threadIdx.x; t < 512; t += 256) {
      int g = t / half, p = t - g * half;
      int i0 = g * len + p, i1 = i0 + half;
      float sa, ca; __sincosf(ang0 * p, &sa, &ca);
      float r1 = re[i1], m1 = im[i1];
      float tr = r1 * ca - m1 * sa, ti = r1 * sa + m1 * ca;
      re[i1] = re[i0] - tr; im[i1] = im[i0] - ti;
      re[i0] += tr;         im[i0] += ti;
    }
    __syncthreads();
  }
  for (int i = threadIdx.x; i < 1024; i += 256) { xr[base + i] = re[i]; xi[base + i] = im[i]; }
}

// ---------------- 3) Patch gather + projection GEMM -> z (B,PN,C,H) ----------------
__global__ void k_proj(const float* __restrict__ xr, const float* __restrict__ xi,
                       const float* __restrict__ pw, const float* __restrict__ pb,
                       float* __restrict__ z) {
  int bp = blockIdx.x; int b = bp / PN_, pn = bp % PN_;
  int wave = threadIdx.x >> 5, lane = threadIdx.x & 31;
  int arow = lane & 15, kh = (lane >> 4) << 1, nc = lane & 15, mb = (lane >> 4) << 3;
  long zbase = (long)bp * C_ * H_;
  for (int t = 0; t < 4; ++t) {
    int tt = wave + (t << 3);
    int m0 = (tt & 1) << 4, n0 = (tt >> 1) << 4;
    long rbase = ((long)b * C_ + (m0 + arow)) * L_ + pn * S_;
    v8f acc = {};
    for (int k = 0; k < 32; k += 4) {
      int k0 = k + kh, k1 = k0 + 1;
      float a0 = (k0 < 16) ? xr[rbase + k0] : xi[rbase + k0 - 16];
      float a1 = (k1 < 16) ? xr[rbase + k1] : xi[rbase + k1 - 16];
      v2f av = { a0, a1 };
      v2f bv = { pw[(long)k0 * H_ + n0 + nc], pw[(long)k1 * H_ + n0 + nc] };
      acc = wmma4(av, bv, acc);
    }
#pragma unroll
    for (int r = 0; r < 8; ++r) {
      int m = m0 + mb + r, n = n0 + nc;
      z[zbase + (long)m * H_ + n] = acc[r] + pb[n];
    }
  }
}

// ---------------- 4) Reduce mg_w2 over last C: W'(H x 32), b'(32) ----------------
__global__ void k_w2red(const float* __restrict__ w2, const float* __restrict__ b2,
                        float* __restrict__ wr, float* __restrict__ br, int l) {
  int idx = blockIdx.x * 256 + threadIdx.x;
  if (idx < H_ * C_) {
    int k = idx >> 5, c2 = idx & 31;
    const float* p = w2 + ((long)l * H_ + k) * (C_ * C_) + c2 * C_;
    float s = 0.f;
    for (int j = 0; j < C_; ++j) s += p[j];
    wr[idx] = s * (1.0f / C_);
  }
  if (blockIdx.x == 0 && threadIdx.x < C_) {
    const float* p = b2 + (long)l * (C_ * C_) + threadIdx.x * C_;
    float s = 0.f;
    for (int j = 0; j < C_; ++j) s += p[j];
    br[threadIdx.x] = s * (1.0f / C_);
  }
}

// ---------------- 5) Mask generator (LDS-staged z, fused GEMMs + gumbel) ----------------
__global__ void k_mask(const float* __restrict__ z, const float* __restrict__ w1,
                       const float* __restrict__ b1, const float* __restrict__ wr,
                       const float* __restrict__ br, float* __restrict__ masks, int l) {
  __shared__ __align__(16) float sZ[C_ * H_];   // 32KB staged z tile
  __shared__ __align__(16) float sH[C_ * H_];   // 32KB relu(z@W1+b1)
  int bp = blockIdx.x;
  long zbase = (long)bp * C_ * H_;
  int tid = threadIdx.x;
  int wave = tid >> 5, lane = tid & 31;
  int nc = lane & 15, mb = (lane >> 4) << 3;
  stage_tile_8k(z + zbase, sZ, tid);
  __syncthreads();
  const float* W1 = w1 + (long)l * H_ * H_;
  for (int t = 0; t < 4; ++t) {
    int tt = wave + (t << 3);
    int m0 = (tt & 1) << 4, n0 = (tt >> 1) << 4;
    v8f acc = {};
    acc = tile_gemm(sZ + m0 * H_, H_, W1 + n0, H_, H_, acc);
#pragma unroll
    for (int r = 0; r < 8; ++r) {
      int m = m0 + mb + r, n = n0 + nc;
      sH[m * H_ + n] = fmaxf(acc[r] + b1[l * H_ + n], 0.f);
    }
  }
  __syncthreads();
  if (wave < 4) {
    int m0 = (wave & 1) << 4, n0 = (wave >> 1) << 4;
    v8f acc = {};
    acc = tile_gemm(sH + m0 * H_, H_, wr + n0, C_, H_, acc);
    long mbase = (long)l * B_ * PN_ * C_ * C_ + (long)bp * C_ * C_;
#pragma unroll
    for (int r = 0; r < 8; ++r) {
      int c1 = m0 + mb + r, c2 = n0 + nc;
      float logit = acc[r] + br[c2];
      float prob = 1.0f / (1.0f + __expf(-logit));
      float u = hash_u((unsigned)(l + 1), (unsigned)(bp * 1024 + c1 * 32 + c2));
      float g = -logf(-logf(u + 1e-10f) + 1e-10f);
      float hard = (prob + g > 0.f) ? 1.f : 0.f;
      masks[mbase + c1 * C_ + c2] = (c1 == c2) ? 1.f : hard;
    }
  }
}

// ---------------- 6) Fused LN -> QKV -> attention -> AV -> +z ----------------
__global__ void k_attn(const float* __restrict__ z, const float* __restrict__ masks,
                       const float* __restrict__ lnw, const float* __restrict__ lnb,
                       const float* __restrict__ wq, const float* __restrict__ bq,
                       const float* __restrict__ wk, const float* __restrict__ bk,
                       const float* __restrict__ wv, const float* __restrict__ bv,
                       float* __restrict__ attnOut, float* __restrict__ maskedOut,
                       float* __restrict__ ao, int l) {
  __shared__ float sQ[C_ * H_], sK[C_ * H_], sV[C_ * H_];  // 3 x 32KB
  __shared__ float sZA[C_ * H_];                            // zn, then 8 x 32x32 scores
  __shared__ float sM[C_ * C_];
  __shared__ float rS[256], rQd[256];
  int bp = blockIdx.x;
  long zbase = (long)bp * C_ * H_;
  int tid = threadIdx.x;
  int wave = tid >> 5, lane = tid & 31;
  int nc = lane & 15, mb = (lane >> 4) << 3;

  // --- Phase A: layernorm over H into sZA ---
  {
    int row = tid >> 3, sub = tid & 7;
    const float* zr = z + zbase + (long)row * H_;
    float s = 0.f, q = 0.f;
    for (int j = sub; j < H_; j += 8) { float v = zr[j]; s += v; q += v * v; }
    rS[tid] = s; rQd[tid] = q; __syncthreads();
    float rs = 0.f, rq2 = 0.f;
    for (int j = 0; j < 8; ++j) { rs += rS[(row << 3) + j]; rq2 += rQd[(row << 3) + j]; }
    float mean = rs * (1.0f / H_);
    float var  = rq2 * (1.0f / H_) - mean * mean;
    float rstd = rsqrtf(var + 1e-5f);
    for (int j = sub; j < H_; j += 8)
      sZA[row * H_ + j] = (zr[j] - mean) * rstd * lnw[l * H_ + j] + lnb[l * H_ + j];
    __syncthreads();
  }

  // --- Phase B: QKV GEMMs from LDS zn ---
  {
    const float* Wm[3] = { wq + (long)l * H_ * H_, wk + (long)l * H_ * H_, wv + (long)l * H_ * H_ };
    const float* Bm[3] = { bq + l * H_, bk + l * H_, bv + l * H_ };
    float* Dm[3] = { sQ, sK, sV };
    for (int qk = 0; qk < 3; ++qk) {
      for (int t = 0; t < 4; ++t) {
        int tt = wave + (t << 3);
        int m0 = (tt & 1) << 4, n0 = (tt >> 1) << 4;
        v8f acc = {};
        acc = tile_gemm(sZA + m0 * H_, H_, Wm[qk] + n0, H_, H_, acc);
#pragma unroll
        for (int r = 0; r < 8; ++r) {
          int m = m0 + mb + r, n = n0 + nc;
          Dm[qk][m * H_ + n] = acc[r] + Bm[qk][n];
        }
      }
    }
    __syncthreads();
  }

  // --- Phase C: per-head scores S = Q K^T * scale, into sZA ---
  int h = wave;
  {
    const float scale = 0.17677669529663687f; // 1/sqrt(32)
    for (int tt = 0; tt < 4; ++tt) {
      int m0 = (tt & 1) << 4, n0 = (tt >> 1) << 4;
      v8f acc = {};
      acc = tile_gemm_bt(sQ + m0 * H_ + h * HD_, H_, sK + n0 * H_ + h * HD_, H_, HD_, acc);
#pragma unroll
      for (int r = 0; r < 8; ++r) {
        int m = m0 + mb + r, n = n0 + nc;
        sZA[h * 1024 + m * 32 + n] = acc[r] * scale;
      }
    }
    __syncthreads();
  }

  // --- Phase D: head means + masked means -> outputs; cache mask in sM ---
  {
    long obase = (long)l * B_ * PN_ * C_ * C_ + (long)bp * C_ * C_;
    for (int e = tid; e < C_ * C_; e += 256) {
      float s = 0.f;
      for (int hh = 0; hh < NH_; ++hh) s += sZA[hh * 1024 + e];
      s *= (1.0f / NH_);
      float mval = masks[obase + e];
      sM[e] = mval;
      attnOut[obase + e]   = s;
      maskedOut[obase + e] = s * mval + (1.f - mval) * (-1e9f);
    }
    __syncthreads();
  }

  // --- Phase E: per-head masked softmax (lane = row) ---
  {
    float* arow = sZA + h * 1024 + lane * 32;
    float vals[32];
    float mx = -3.4e38f;
#pragma unroll
    for (int j = 0; j < 32; ++j) {
      float mval = sM[lane * 32 + j];
      float xv = arow[j] * mval + (1.f - mval) * (-1e9f);
      vals[j] = xv; mx = fmaxf(mx, xv);
    }
    float sum = 0.f;
#pragma unroll
    for (int j = 0; j < 32; ++j) { vals[j] = __expf(vals[j] - mx); sum += vals[j]; }
    float inv = 1.0f / sum;
#pragma unroll
    for (int j = 0; j < 32; ++j) arow[j] = vals[j] * inv;
    __syncthreads();
  }

  // --- Phase F: ao = aw @ V_h + z ---
  for (int tt = 0; tt < 4; ++tt) {
    int m0 = (tt & 1) << 4, n0 = (tt >> 1) << 4;
    v8f acc = {};
    acc = tile_gemm(sZA + h * 1024 + m0 * 32, 32, sV + h * HD_ + n0, H_, HD_, acc);
#pragma unroll
    for (int r = 0; r < 8; ++r) {
      int m = m0 + mb + r;
      int col = h * HD_ + n0 + nc;
      ao[zbase + (long)m * H_ + col] = acc[r] + z[zbase + (long)m * H_ + col];
    }
  }
}

// ---------------- 7) Fused FFN: z = gelu(ao@W1+b1)@W2 + b2 + ao ----------------
__global__ void k_ffn(const float* __restrict__ ao, const float* __restrict__ w1,
                      const float* __restrict__ b1, const float* __restrict__ w2,
                      const float* __restrict__ b2, float* __restrict__ z, int l) {
  __shared__ __align__(16) float sF[C_ * 512];  // 64KB hidden chunk
  __shared__ __align__(16) float sA[C_ * H_];   // 32KB activation tile
  int bp = blockIdx.x;
  long base = (long)bp * C_ * H_;
  int tid = threadIdx.x, wave = tid >> 5, lane = tid & 31;
  int nc = lane & 15, mb = (lane >> 4) << 3;
  stage_tile_8k(ao + base, sA, tid);
  __syncthreads();
  v8f accO[4] = {};
  const float* W1 = w1 + (long)l * H_ * FH_;
  const float* W2 = w2 + (long)l * FH_ * H_;
  for (int cc = 0; cc < 2; ++cc) {
    for (int t = 0; t < 8; ++t) {
      int tt = wave + (t << 3);                 // 0..63
      int m0 = (tt & 1) << 4, n0 = (tt >> 1) << 4; // n0: 0..496
      v8f acc = {};
      acc = tile_gemm(sA + m0 * H_, H_, W1 + cc * 512 + n0, FH_, H_, acc);
#pragma unroll
      for (int r = 0; r < 8; ++r) {
        int m = m0 + mb + r, n = n0 + nc;
        float xg = acc[r] + b1[l * FH_ + cc * 512 + n];
        sF[m * 512 + n] = geluf(xg);
      }
    }
    __syncthreads();
    for (int t = 0; t < 4; ++t) {
      int tt = wave + (t << 3);
      int m0 = (tt & 1) << 4, n0 = (tt >> 1) << 4;
      accO[t] = tile_gemm(sF + m0 * 512, 512, W2 + (long)cc * 512 * H_ + n0, H_, 512, accO[t]);
    }
    __syncthreads();
  }
  for (int t = 0; t < 4; ++t) {
    int tt = wave + (t << 3);
    int m0 = (tt & 1) << 4, n0 = (tt >> 1) << 4;
#pragma unroll
    for (int r = 0; r < 8; ++r) {
      int m = m0 + mb + r, n = n0 + nc;
      z[base + (long)m * H_ + n] = accO[t][r] + b2[l * H_ + n] + sA[m * H_ + n];
    }
  }
}

// ---------------- 8) Final projection: block-cooperative LDS panel GEMM ----------------
// Per block: one (b, part, 64-column group). 8 waves = 2(m) x 4(n) 16x16 tiles.
__global__ void k_final(const float* __restrict__ z, const float* __restrict__ prw,
                        const float* __restrict__ prb, const float* __restrict__ piw,
                        const float* __restrict__ pib, float* __restrict__ xrr,
                        float* __restrict__ xir, float* __restrict__ o1,
                        float* __restrict__ o2) {
  const int KTOT = PN_ * (H_ / 2);  // 16256
  __shared__ __align__(16) float sB[64 * 64];   // 16KB B panel [kk][col]
  __shared__ __align__(16) float sA2[32 * 64];  // 8KB  A panel [m][kk]
  int blk = blockIdx.x;             // 512 blocks
  int b = blk >> 5; int rem = blk & 31; int part = rem >> 4; int sub = rem & 15;
  int ncol0 = sub * 64;
  int tid = threadIdx.x;
  int wave = tid >> 5, lane = tid & 31;
  int nc = lane & 15, mb = (lane >> 4) << 3;
  int m0 = (wave & 1) << 4;
  int n0loc = (wave >> 1) << 4;     // 0..48
  const float* W    = part ? piw : prw;
  const float* bias = part ? pib : prb;
  long zb0 = (long)b * PN_ * C_ * H_ + part * (H_ / 2);
  v8f acc = {};
  for (int k0 = 0; k0 < KTOT; k0 += 64) {
    // cooperative A panel: As[m][kk] = z[b, (k0+kk)/128, m, (k0+kk)%128 + part*128]
    for (int e = tid; e < 32 * 64; e += 256) {
      int m = e >> 6, kk = e & 63;
      int kg = k0 + kk;
      sA2[e] = z[zb0 + (long)(kg >> 7) * C_ * H_ + (long)m * H_ + (kg & 127)];
    }
    // cooperative B panel: Bs[kk][col] = W[(k0+kk)*L + ncol0 + col]  (256B rows, coalesced)
    for (int e = tid; e < 64 * 64; e += 256) {
      int kk = e >> 6, col = e & 63;
      sB[e] = W[(long)(k0 + kk) * L_ + ncol0 + col];
    }
    if (tid < 64 && k0 + 64 < KTOT)
      __builtin_prefetch(W + (long)(k0 + 64 + tid) * L_ + ncol0, 0, 1);
    __syncthreads();
    acc = tile_gemm(sA2 + m0 * 64, 64, sB + n0loc, 64, 64, acc);
    __syncthreads();
  }
  float* xbuf = part ? xir : xrr;
  float* obuf = part ? o2 : o1;
#pragma unroll
  for (int r = 0; r < 8; ++r) {
    int m = m0 + mb + r, n = ncol0 + n0loc + nc;
    float v = acc[r] + bias[n];
    xbuf[((long)b * C_ + m) * L_ + n] = v;
    obuf[((long)b * L_ + n) * C_ + m] = v;
  }
}

// ---------------- 9) Inverse FFT -> real part, transposed to (B,L,C) ----------------
__global__ void k_ifft(const float* __restrict__ xrr, const float* __restrict__ xir,
                       float* __restrict__ o0) {
  __shared__ float re[1024], im[1024];
  int bc = blockIdx.x; int b = bc >> 5, c = bc & 31;
  long base = (long)bc * L_;
  for (int i = threadIdx.x; i < 1024; i += 256) {
    int j = __brev((unsigned)i) >> 22;
    re[j] = xrr[base + i]; im[j] = xir[base + i];
  }
  __syncthreads();
  for (int len = 2; len <= 1024; len <<= 1) {
    int half = len >> 1;
    float ang0 = 6.283185307179586f / (float)len;  // + sign for inverse
    for (int t = threadIdx.x; t < 512; t += 256) {
      int g = t / half, p = t - g * half;
      int i0 = g * len + p, i1 = i0 + half;
      float sa, ca; __sincosf(ang0 * p, &sa, &ca);
      float r1 = re[i1], m1 = im[i1];
      float tr = r1 * ca - m1 * sa, ti = r1 * sa + m1 * ca;
      re[i1] = re[i0] - tr; im[i1] = im[i0] - ti;
      re[i0] += tr;         im[i0] += ti;
    }
    __syncthreads();
  }
  for (int i = threadIdx.x; i < 1024; i += 256)
    o0[((long)b * L_ + i) * C_ + c] = re[i] * (1.0f / 1024.0f);
}

// ---------------- launch ----------------
extern "C" void kernel_launch(void* const* d_in, const int* in_sizes, int n_in,
                              void* d_out, int out_size, void* d_ws, size_t ws_size,
                              hipStream_t stream) {
  const float* x      = (const float*)d_in[0];
  const float* in_w   = (const float*)d_in[1];
  const float* in_b   = (const float*)d_in[2];
  const float* proj_w = (const float*)d_in[3];
  const float* proj_b = (const float*)d_in[4];
  const float* mg_w1  = (const float*)d_in[5];
  const float* mg_b1  = (const float*)d_in[6];
  const float* mg_w2  = (const float*)d_in[7];
  const float* mg_b2  = (const float*)d_in[8];
  const float* ln_w   = (const float*)d_in[9];
  const float* ln_b   = (const float*)d_in[10];
  const float* wq     = (const float*)d_in[11];
  const float* bq     = (const float*)d_in[12];
  const float* wk     = (const float*)d_in[13];
  const float* bk     = (const float*)d_in[14];
  const float* wv     = (const float*)d_in[15];
  const float* bv     = (const float*)d_in[16];
  const float* ffn_w1 = (const float*)d_in[17];
  const float* ffn_b1 = (const float*)d_in[18];
  const float* ffn_w2 = (const float*)d_in[19];
  const float* ffn_b2 = (const float*)d_in[20];
  const float* pr_w   = (const float*)d_in[21];
  const float* pr_b   = (const float*)d_in[22];
  const float* pi_w   = (const float*)d_in[23];
  const float* pi_b   = (const float*)d_in[24];

  float* out = (float*)d_out;
  const long NBLC = (long)B_ * L_ * C_;             // 524288
  const long NMSK = (long)NL_ * B_ * PN_ * C_ * C_; // 4161536
  float* o_xrec = out;
  float* o_xr   = out + NBLC;
  float* o_xi   = out + 2 * NBLC;
  float* o_mask = out + 3 * NBLC;
  float* o_attn = o_mask + NMSK;
  float* o_msk  = o_attn + NMSK;

  float* w = (float*)d_ws;
  const long NZ = (long)B_ * PN_ * C_ * H_;         // 16646144
  float* xn  = w;                 // 524288
  float* xr  = xn  + NBLC;        // 524288
  float* xi  = xr  + NBLC;        // 524288
  float* z   = xi  + NBLC;        // NZ
  float* ao  = z   + NZ;          // NZ
  float* w2r = ao  + NZ;          // 8192
  float* w2rb = w2r + H_ * C_;    // 64 (padded)
  float* xrr = w2rb + 64;         // 524288
  float* xir = xrr + NBLC;        // 524288

  const int BP = B_ * PN_;        // 2032

  k_ln_time<<<B_ * C_, 256, 0, stream>>>(x, in_w, in_b, xn);
  k_fft_fwd<<<B_ * C_, 256, 0, stream>>>(xn, xr, xi);
  k_proj<<<BP, 256, 0, stream>>>(xr, xi, proj_w, proj_b, z);

  for (int l = 0; l < NL_; ++l) {
    k_w2red<<<32, 256, 0, stream>>>(mg_w2, mg_b2, w2r, w2rb, l);
    k_mask<<<BP, 256, 0, stream>>>(z, mg_w1, mg_b1, w2r, w2rb, o_mask, l);
    k_attn<<<BP, 256, 0, stream>>>(z, o_mask, ln_w, ln_b, wq, bq, wk, bk, wv, bv,
                                   o_attn, o_msk, ao, l);
    k_ffn<<<BP, 256, 0, stream>>>(ao, ffn_w1, ffn_b1, ffn_w2, ffn_b2, z, l);
  }

  k_final<<<512, 256, 0, stream>>>(z, pr_w, pr_b, pi_w, pi_b, xrr, xir, o_xr, o_xi);
  k_ifft<<<B_ * C_, 256, 0, stream>>>(xrr, xir, o_xrec);
}